// DGCNConv_85452669321970
// MI455X (gfx1250) — compile-verified
//
#include <hip/hip_runtime.h>
#include <math.h>

// Problem constants (match reference setup_inputs()).
#define Nn   8192      // nodes
#define Hh   128       // feature/hidden dim
#define Ll   4         // layers
#define Gg   32        // graphs
#define NPGc 256       // nodes per graph
#define Kk   60        // sortpool k
#define Dd   (Ll*Hh)   // concat feature dim = 512

#define BM 64          // rows per block tile
#define BK 32          // k-chunk
#define PAD_A 4        // As row stride 36 floats (144B, 16B aligned, odd bank step)

typedef float v2f __attribute__((ext_vector_type(2)));
typedef float v8f __attribute__((ext_vector_type(8)));
typedef unsigned int v4u __attribute__((ext_vector_type(4)));
typedef int v4i __attribute__((ext_vector_type(4)));
typedef int v8i __attribute__((ext_vector_type(8)));
typedef __attribute__((address_space(3))) float lds_f;

// ---------------------------------------------------------------- TDM helper
// Issue a TDM 2D tile load: BM x BK f32 tile of an Nn-wide row-major tensor,
// padded +4 DWORDs every 32 DWORDs so LDS rows land on a 36-float stride.
__device__ __forceinline__ void tdm_load_tile(const float* gaddr, unsigned lds_off,
                                              v8i g1) {
  unsigned long long ga = (unsigned long long)gaddr;
  v4u g0;
  g0.x = 1u;                                   // count=1, user descriptor
  g0.y = lds_off;                              // lds_addr (bytes)
  g0.z = (unsigned)ga;                         // global_addr[31:0]
  g0.w = (unsigned)((ga >> 32) & 0x01FFFFFFu)  // global_addr[56:32]
         | 0x80000000u;                        // type=2 ("image")
  v4i z4 = {0, 0, 0, 0};
#if __clang_major__ >= 23
  v8i z8 = {0, 0, 0, 0, 0, 0, 0, 0};
  __builtin_amdgcn_tensor_load_to_lds(g0, g1, z4, z4, z8, 0);
#else
  __builtin_amdgcn_tensor_load_to_lds(g0, g1, z4, z4, 0);
#endif
}

__device__ __forceinline__ v8i tdm_group1_desc() {
  // data_size=4B (2<<16) | pad_enable (1<<20) | pad_interval=32DW (4<<22)
  // | pad_amount=4DW (3<<25)
  v8i g1;
  g1[0] = 0x07120000;          // mask/flags/padding, workgroup_mask=0
  g1[1] = 0x20000000;          // tensor_dim0[15:0]=8192 at bits [63:48]
  g1[2] = 0x20000000;          // tensor_dim1[15:0]=8192 at bits [95:80]
  g1[3] = (BK << 16);          // tile_dim0=32 at bits [127:112]
  g1[4] = BM;                  // tile_dim1=64 at bits [143:128]
  g1[5] = Nn;                  // tensor_dim0_stride[31:0]=8192
  g1[6] = 0;
  g1[7] = 0;
  return g1;
}

// ---------------------------------------------------------------- utilities

__global__ void fill_zero_f4(float4* __restrict__ p, long n4) {
  long i = (long)blockIdx.x * blockDim.x + threadIdx.x;
  if (i < n4) p[i] = make_float4(0.f, 0.f, 0.f, 0.f);
}

// Scatter edges into dense symmetrized adjacency + degree counters.
__global__ void edge_accum(const int* __restrict__ ei, int E,
                           float* __restrict__ AT,
                           float* __restrict__ degi, float* __restrict__ dego) {
  int e = blockIdx.x * blockDim.x + threadIdx.x;
  if (e >= E) return;
  int u = ei[e];        // row (source)
  int v = ei[E + e];    // col (dest)
  atomicAdd(&AT[(size_t)u * Nn + v], 1.0f);   // A contribution
  atomicAdd(&AT[(size_t)v * Nn + u], 1.0f);   // A^T contribution
  atomicAdd(&dego[u], 1.0f);                  // row sum of A
  atomicAdd(&degi[v], 1.0f);                  // col sum of A
}

// Self loops (+2 on diagonal of A+A^T, +1 to each degree) then deg^-1/2.
__global__ void finalize_deg(float* __restrict__ AT,
                             float* __restrict__ degi, float* __restrict__ dego) {
  int i = blockIdx.x * blockDim.x + threadIdx.x;
  if (i >= Nn) return;
  AT[(size_t)i * Nn + i] += 2.0f;
  degi[i] = rsqrtf(degi[i] + 1.0f);
  dego[i] = rsqrtf(dego[i] + 1.0f);
}

// y[i,f] = s[i] * x[i,f]  (fold D^-1/2 column scaling of A into X)
__global__ void colscale(const float4* __restrict__ x, const float* __restrict__ s,
                         float4* __restrict__ y) {
  int i = blockIdx.x * blockDim.x + threadIdx.x;
  if (i >= Nn * Hh / 4) return;
  float sc = s[i / (Hh / 4)];
  float4 v = x[i];
  y[i] = make_float4(sc * v.x, sc * v.y, sc * v.z, sc * v.w);
}

// dst[r, doff + c] = 0.5*(0.1*a[r,c] + 0.9*b[r,c]), dst row stride = dstride
__global__ void combine(const float4* __restrict__ a, const float4* __restrict__ b,
                        float* __restrict__ dst, int dstride, int doff) {
  int i = blockIdx.x * blockDim.x + threadIdx.x;
  if (i >= Nn * Hh / 4) return;
  int r = i / (Hh / 4), c4 = i % (Hh / 4);
  float4 va = a[i], vb = b[i];
  float4 o = make_float4(0.5f * (0.1f * va.x + 0.9f * vb.x),
                         0.5f * (0.1f * va.y + 0.9f * vb.y),
                         0.5f * (0.1f * va.z + 0.9f * vb.z),
                         0.5f * (0.1f * va.w + 0.9f * vb.w));
  *(float4*)(dst + (size_t)r * dstride + doff + c4 * 4) = o;
}

// ------------------------------------------------- dense A_T @ X with WMMA f32
// Y[i,:] = rs[i] * sum_k A[i,k] * X[k,:]    (A: NxN f32, X: NxH, Y: NxH)
// A tiles arrive via TDM (double buffered, pad -> 36-float LDS rows).
// X tiles staged K-pair-interleaved so B fragments are single ds_load_b64.
__global__ __launch_bounds__(256)
void spmm_wmma(const float* __restrict__ A, const float* __restrict__ X,
               const float* __restrict__ rs, float* __restrict__ Y) {
  __shared__ float As[2][BM][BK + PAD_A];      // 2 x 64 x 36 (TDM destination)
  __shared__ float BsI[BK / 2][Hh][2];         // (k/2, n, k&1) interleaved

  const int t    = threadIdx.x;
  const int lane = t & 31;
  const int wave = t >> 5;
  const int half = lane >> 4;     // 0: lanes 0-15, 1: lanes 16-31
  const int lid  = lane & 15;
  const int wr   = wave & 3;      // 16-row strip within 64-row tile
  const int wc   = wave >> 2;     // 64-col strip within 128 cols
  const int row0 = blockIdx.x * BM;

  const unsigned ldsA0 =
      (unsigned)(unsigned long long)(lds_f*)&As[0][0][0];
  const unsigned abytes = (unsigned)(BM * (BK + PAD_A) * sizeof(float));
  const v8i g1 = tdm_group1_desc();

  v8f c[4];
#pragma unroll
  for (int nt = 0; nt < 4; ++nt)
#pragma unroll
    for (int r = 0; r < 8; ++r) c[nt][r] = 0.0f;

  // Prime the pipeline: tile k0=0 -> buffer 0.
  if (wave == 0) tdm_load_tile(A + (size_t)row0 * Nn, ldsA0, g1);

  for (int k0 = 0; k0 < Nn; k0 += BK) {
    const int buf = (k0 >> 5) & 1;

    // Prefetch next A tile into the other buffer while we stage X + compute.
    if (wave == 0 && k0 + BK < Nn)
      tdm_load_tile(A + (size_t)row0 * Nn + (k0 + BK), ldsA0 + (buf ^ 1) * abytes, g1);

    // X tile 32x128 -> K-pair interleaved LDS (1024 float4, 4 per thread)
#pragma unroll
    for (int j = 0; j < 4; ++j) {
      int idx4 = t + j * 256;
      int r = idx4 >> 5, c4 = idx4 & 31;
      float4 v = *(const float4*)(X + (size_t)(k0 + r) * Hh + c4 * 4);
      int p = r >> 1, i = r & 1;
      BsI[p][c4 * 4 + 0][i] = v.x;
      BsI[p][c4 * 4 + 1][i] = v.y;
      BsI[p][c4 * 4 + 2][i] = v.z;
      BsI[p][c4 * 4 + 3][i] = v.w;
    }

    // Current A tile must have landed before anyone reads it.
    if (wave == 0) {
      if (k0 + BK < Nn) __builtin_amdgcn_s_wait_tensorcnt(1);
      else              __builtin_amdgcn_s_wait_tensorcnt(0);
    }
    __syncthreads();

#pragma unroll
    for (int kk = 0; kk < BK; kk += 4) {
      // A frag per ISA 7.12.2: lanes 0-15 hold K=2*half, K=2*half+1
      v2f a;
      a.x = As[buf][wr * 16 + lid][kk + 2 * half];
      a.y = As[buf][wr * 16 + lid][kk + 2 * half + 1];
#pragma unroll
      for (int nt = 0; nt < 4; ++nt) {
        v2f b = *(const v2f*)&BsI[(kk >> 1) + half][wc * 64 + nt * 16 + lid][0];
        c[nt] = __builtin_amdgcn_wmma_f32_16x16x4_f32(
            false, a, false, b, (short)0, c[nt], false, false);
      }
    }
    __syncthreads();
  }

  // Epilogue: row scaling by rs[row]; C/D layout: M = r + 8*half, N = lid.
#pragma unroll
  for (int r = 0; r < 8; ++r) {
    int row = row0 + wr * 16 + r + 8 * half;
    float sc = rs[row];
#pragma unroll
    for (int nt = 0; nt < 4; ++nt) {
      int col = wc * 64 + nt * 16 + lid;
      Y[(size_t)row * Hh + col] = sc * c[nt][r];
    }
  }
}

// ----------------------------------------- Y = tanh(T @ W^T + bias), WMMA f32
__global__ __launch_bounds__(256)
void gemm2_tanh(const float* __restrict__ T, const float* __restrict__ W,
                const float* __restrict__ bias, float* __restrict__ Y) {
  __shared__ float As[BM][BK + PAD_A];
  __shared__ float BsI[BK / 2][Hh][2];   // (k/2, o, k&1): BsI[..] = W[o*H + k]
  __shared__ float bsh[Hh];

  const int t    = threadIdx.x;
  const int lane = t & 31;
  const int wave = t >> 5;
  const int half = lane >> 4;
  const int lid  = lane & 15;
  const int wr   = wave & 3;
  const int wc   = wave >> 2;
  const int row0 = blockIdx.x * BM;

  if (t < Hh) bsh[t] = bias[t];

  v8f c[4];
#pragma unroll
  for (int nt = 0; nt < 4; ++nt)
#pragma unroll
    for (int r = 0; r < 8; ++r) c[nt][r] = 0.0f;

  for (int k0 = 0; k0 < Hh; k0 += BK) {
    // T tile 64x32 (512 float4, 2 per thread)
#pragma unroll
    for (int j = 0; j < 2; ++j) {
      int idx4 = t + j * 256;
      int r = idx4 >> 3, c4 = idx4 & 7;
      float4 v = *(const float4*)(T + (size_t)(row0 + r) * Hh + k0 + c4 * 4);
      *(float4*)(&As[r][c4 * 4]) = v;
    }
    // W^T tile: coalesced read of W rows; thread holds 4 consecutive K for one
    // output column o -> two contiguous (K,K+1) pairs -> b64 LDS stores.
#pragma unroll
    for (int j = 0; j < 4; ++j) {
      int idx4 = t + j * 256;          // 1024 float4
      int o = idx4 >> 3, k4 = idx4 & 7;
      float4 w = *(const float4*)(W + (size_t)o * Hh + k0 + k4 * 4);
      v2f lo, hi;
      lo.x = w.x; lo.y = w.y;
      hi.x = w.z; hi.y = w.w;
      *(v2f*)&BsI[k4 * 2 + 0][o][0] = lo;
      *(v2f*)&BsI[k4 * 2 + 1][o][0] = hi;
    }
    __syncthreads();

#pragma unroll
    for (int kk = 0; kk < BK; kk += 4) {
      v2f a;
      a.x = As[wr * 16 + lid][kk + 2 * half];
      a.y = As[wr * 16 + lid][kk + 2 * half + 1];
#pragma unroll
      for (int nt = 0; nt < 4; ++nt) {
        v2f b = *(const v2f*)&BsI[(kk >> 1) + half][wc * 64 + nt * 16 + lid][0];
        c[nt] = __builtin_amdgcn_wmma_f32_16x16x4_f32(
            false, a, false, b, (short)0, c[nt], false, false);
      }
    }
    __syncthreads();
  }

#pragma unroll
  for (int r = 0; r < 8; ++r) {
    int row = row0 + wr * 16 + r + 8 * half;
#pragma unroll
    for (int nt = 0; nt < 4; ++nt) {
      int col = wc * 64 + nt * 16 + lid;
      Y[(size_t)row * Hh + col] = tanhf(c[nt][r] + bsh[col]);
    }
  }
}

// -------------------------------------------------------- sortpool (top-K=60)
__global__ __launch_bounds__(256)
void sortpool(const float* __restrict__ feat, float* __restrict__ out) {
  __shared__ float vals[NPGc];
  __shared__ int   sel[Kk];
  int g = blockIdx.x;
  int t = threadIdx.x;
  const float* fg = feat + (size_t)g * NPGc * Dd;

  float v = fg[(size_t)t * Dd + (Dd - 1)];
  vals[t] = v;
  __syncthreads();

  // Stable descending rank (ties -> lower index first), matches lax.top_k.
  int rank = 0;
  for (int j = 0; j < NPGc; ++j) {
    float vj = vals[j];
    rank += (vj > v) || (vj == v && j < t);
  }
  if (rank < Kk) sel[rank] = t;
  __syncthreads();

  for (int i = t; i < Kk * Dd; i += 256) {
    int r = i / Dd, col = i - r * Dd;
    out[(size_t)g * Kk * Dd + i] = fg[(size_t)sel[r] * Dd + col];
  }
}

// ---------------------------------------------------------------- launcher

extern "C" void kernel_launch(void* const* d_in, const int* in_sizes, int n_in,
                              void* d_out, int out_size, void* d_ws, size_t ws_size,
                              hipStream_t stream) {
  const float* x  = (const float*)d_in[0];
  const int*   ei = (const int*)d_in[1];
  const float* W1 = (const float*)d_in[2];
  const float* b1 = (const float*)d_in[3];
  const float* W2 = (const float*)d_in[4];
  const float* b2 = (const float*)d_in[5];
  const float* W3 = (const float*)d_in[6];
  const float* b3 = (const float*)d_in[7];
  const int E = in_sizes[1] / 2;
  float* out = (float*)d_out;

  // Workspace layout (floats). Needs ~315 MB.
  float* ws = (float*)d_ws;
  size_t off = 0;
  float* AT   = ws + off; off += (size_t)Nn * Nn;   // dense A + A^T
  float* degi = ws + off; off += Nn;                // d_in^-1/2 (after finalize)
  float* dego = ws + off; off += Nn;                // d_out^-1/2
  float* cur1 = ws + off; off += (size_t)Nn * Hh;
  float* cur2 = ws + off; off += (size_t)Nn * Hh;
  float* mid  = ws + off; off += (size_t)Nn * Hh;
  float* xs   = ws + off; off += (size_t)Nn * Hh;   // column-scaled X
  float* tbuf = ws + off; off += (size_t)Nn * Hh;   // A_norm @ X result
  float* y3   = ws + off; off += (size_t)Nn * Hh;
  float* y4   = ws + off; off += (size_t)Nn * Hh;
  float* feat = ws + off; off += (size_t)Nn * Dd;   // [N, 512]

  const int ethreads = 256;
  const long at4 = (long)Nn * Nn / 4;
  fill_zero_f4<<<(unsigned)((at4 + 255) / 256), ethreads, 0, stream>>>((float4*)AT, at4);
  const long dg4 = (long)(2 * Nn) / 4;  // degi & dego are contiguous
  fill_zero_f4<<<(unsigned)((dg4 + 255) / 256), ethreads, 0, stream>>>((float4*)degi, dg4);

  edge_accum<<<(E + 255) / 256, ethreads, 0, stream>>>(ei, E, AT, degi, dego);
  finalize_deg<<<Nn / 256, ethreads, 0, stream>>>(AT, degi, dego);

  hipMemcpyAsync(cur1, x, (size_t)Nn * Hh * sizeof(float), hipMemcpyDeviceToDevice, stream);
  hipMemcpyAsync(cur2, x, (size_t)Nn * Hh * sizeof(float), hipMemcpyDeviceToDevice, stream);

  const int nEl4 = Nn * Hh / 4;
  const int ewBlocks = (nEl4 + 255) / 256;
  const int gemmBlocks = Nn / BM;   // 128

  for (int lv = 0; lv < Ll; ++lv) {
    const float* W1l = W1 + (size_t)lv * Hh * Hh;
    const float* b1l = b1 + (size_t)lv * Hh;
    const float* W2l = W2 + (size_t)lv * Hh * Hh;
    const float* b2l = b2 + (size_t)lv * Hh;
    const float* W3l = W3 + (size_t)lv * Hh * Hh;
    const float* b3l = b3 + (size_t)lv * Hh;

    // cur1 = tanh((A1 @ cur1) @ W1^T + b1);   A1 = din . AT . din
    colscale<<<ewBlocks, ethreads, 0, stream>>>((const float4*)cur1, degi, (float4*)xs);
    spmm_wmma<<<gemmBlocks, ethreads, 0, stream>>>(AT, xs, degi, tbuf);
    gemm2_tanh<<<gemmBlocks, ethreads, 0, stream>>>(tbuf, W1l, b1l, cur1);

    // cur2 = tanh((A2 @ cur2) @ W2^T + b2);   A2 = dout . AT . dout
    colscale<<<ewBlocks, ethreads, 0, stream>>>((const float4*)cur2, dego, (float4*)xs);
    spmm_wmma<<<gemmBlocks, ethreads, 0, stream>>>(AT, xs, dego, tbuf);
    gemm2_tanh<<<gemmBlocks, ethreads, 0, stream>>>(tbuf, W2l, b2l, cur2);

    // mid = 0.5*(0.1*cur1 + 0.9*cur2)
    combine<<<ewBlocks, ethreads, 0, stream>>>((const float4*)cur1, (const float4*)cur2,
                                               mid, Hh, 0);

    // c3 = tanh((A1 @ mid) @ W3^T + b3)
    colscale<<<ewBlocks, ethreads, 0, stream>>>((const float4*)mid, degi, (float4*)xs);
    spmm_wmma<<<gemmBlocks, ethreads, 0, stream>>>(AT, xs, degi, tbuf);
    gemm2_tanh<<<gemmBlocks, ethreads, 0, stream>>>(tbuf, W3l, b3l, y3);

    // c4 = tanh((A2 @ mid) @ W3^T + b3)
    colscale<<<ewBlocks, ethreads, 0, stream>>>((const float4*)mid, dego, (float4*)xs);
    spmm_wmma<<<gemmBlocks, ethreads, 0, stream>>>(AT, xs, dego, tbuf);
    gemm2_tanh<<<gemmBlocks, ethreads, 0, stream>>>(tbuf, W3l, b3l, y4);

    // feat[:, lv*128:(lv+1)*128] = 0.5*(0.1*c3 + 0.9*c4)
    combine<<<ewBlocks, ethreads, 0, stream>>>((const float4*)y3, (const float4*)y4,
                                               feat, Dd, lv * Hh);
  }

  sortpool<<<Gg, ethreads, 0, stream>>>(feat, out);
}